// MultiBoxLoss_73735998538406
// MI455X (gfx1250) — compile-verified
//
#include <hip/hip_runtime.h>
#include <hip/hip_bf16.h>

typedef float v2f __attribute__((ext_vector_type(2)));
typedef float v8f __attribute__((ext_vector_type(8)));

#define MAXO 16

// ---------------------------------------------------------------------------
// Kernel 1: per-batch matching + encode + smooth-L1 (fused; loc_t never stored)
// One block per batch image. 256 threads = 8 wave32s.
// ---------------------------------------------------------------------------
__global__ __launch_bounds__(256) void match_kernel(
    const float* __restrict__ loc, const float* __restrict__ priors,
    const float* __restrict__ targets, int* __restrict__ conf_t,
    int* __restrict__ num_pos, float* __restrict__ lossl_part,
    int P, int O)
{
    const int b = blockIdx.x, tid = threadIdx.x;
    __shared__ float tr[MAXO][4];
    __shared__ float tlab[MAXO];
    __shared__ float tarea[MAXO];
    __shared__ float redv[256];
    __shared__ int   redi[256];
    __shared__ int   bpi[MAXO];

    if (tid < O) {
        const float* t = targets + ((size_t)b * O + tid) * 5;
        float x1 = t[0], y1 = t[1], x2 = t[2], y2 = t[3];
        tr[tid][0] = x1; tr[tid][1] = y1; tr[tid][2] = x2; tr[tid][3] = y2;
        tlab[tid] = t[4];
        tarea[tid] = (x2 - x1) * (y2 - y1);
    }
    __syncthreads();

    // -------- Phase A: best prior per truth (argmax over P, first-max wins) --
    float bo[MAXO]; int bi[MAXO];
    for (int o = 0; o < MAXO; ++o) { bo[o] = -1.0f; bi[o] = 0; }

    for (int p = tid; p < P; p += 256) {
        float4 pr = ((const float4*)priors)[p];
        float px1 = pr.x - pr.z * 0.5f, py1 = pr.y - pr.w * 0.5f;
        float px2 = pr.x + pr.z * 0.5f, py2 = pr.y + pr.w * 0.5f;
        float areab = (px2 - px1) * (py2 - py1);   // corner-form area (match ref fp)
        for (int o = 0; o < O; ++o) {
            float ix = fmaxf(fminf(tr[o][2], px2) - fmaxf(tr[o][0], px1), 0.0f);
            float iy = fmaxf(fminf(tr[o][3], py2) - fmaxf(tr[o][1], py1), 0.0f);
            float inter = ix * iy;
            float ov = inter / (tarea[o] + areab - inter);
            if (ov > bo[o]) { bo[o] = ov; bi[o] = p; }   // strict > : first max
        }
    }
    // cross-thread reduce, tie-break to smaller prior index (stable argmax)
    for (int o = 0; o < O; ++o) {
        redv[tid] = bo[o]; redi[tid] = bi[o];
        __syncthreads();
        for (int s = 128; s > 0; s >>= 1) {
            if (tid < s) {
                float v2 = redv[tid + s]; int i2 = redi[tid + s];
                if (v2 > redv[tid] || (v2 == redv[tid] && i2 < redi[tid])) {
                    redv[tid] = v2; redi[tid] = i2;
                }
            }
            __syncthreads();
        }
        if (tid == 0) bpi[o] = redi[0];
        __syncthreads();
    }

    // -------- Phase B: best truth per prior, scatter fix, conf_t, smooth-L1 --
    float sl = 0.0f; int pc = 0;
    for (int p = tid; p < P; p += 256) {
        float4 pr = ((const float4*)priors)[p];
        float px1 = pr.x - pr.z * 0.5f, py1 = pr.y - pr.w * 0.5f;
        float px2 = pr.x + pr.z * 0.5f, py2 = pr.y + pr.w * 0.5f;
        float areab = (px2 - px1) * (py2 - py1);
        float bestv = -1.0f; int bestt = 0;
        for (int o = 0; o < O; ++o) {
            float ix = fmaxf(fminf(tr[o][2], px2) - fmaxf(tr[o][0], px1), 0.0f);
            float iy = fmaxf(fminf(tr[o][3], py2) - fmaxf(tr[o][1], py1), 0.0f);
            float inter = ix * iy;
            float ov = inter / (tarea[o] + areab - inter);
            if (ov > bestv) { bestv = ov; bestt = o; }   // first max over truths
        }
        // ensure every gt keeps its best prior (ascending o -> last wins)
        for (int o = 0; o < O; ++o) {
            if (bpi[o] == p) { bestt = o; bestv = 2.0f; }
        }
        int cf = (bestv < 0.5f) ? 0 : ((int)tlab[bestt] + 1);
        conf_t[(size_t)b * P + p] = cf;
        if (cf > 0) {
            ++pc;
            float mx1 = tr[bestt][0], my1 = tr[bestt][1];
            float mx2 = tr[bestt][2], my2 = tr[bestt][3];
            float t0 = ((mx1 + mx2) * 0.5f - pr.x) / (0.1f * pr.z);
            float t1 = ((my1 + my2) * 0.5f - pr.y) / (0.1f * pr.w);
            float t2 = logf((mx2 - mx1) / pr.z) / 0.2f;
            float t3 = logf((my2 - my1) / pr.w) / 0.2f;
            float4 L = ((const float4*)loc)[(size_t)b * P + p];
            float d0 = fabsf(L.x - t0), d1 = fabsf(L.y - t1);
            float d2 = fabsf(L.z - t2), d3 = fabsf(L.w - t3);
            sl += (d0 < 1.0f ? 0.5f * d0 * d0 : d0 - 0.5f);
            sl += (d1 < 1.0f ? 0.5f * d1 * d1 : d1 - 0.5f);
            sl += (d2 < 1.0f ? 0.5f * d2 * d2 : d2 - 0.5f);
            sl += (d3 < 1.0f ? 0.5f * d3 * d3 : d3 - 0.5f);
        }
    }
    redv[tid] = sl; redi[tid] = pc;
    __syncthreads();
    for (int s = 128; s > 0; s >>= 1) {
        if (tid < s) { redv[tid] += redv[tid + s]; redi[tid] += redi[tid + s]; }
        __syncthreads();
    }
    if (tid == 0) { lossl_part[b] = redv[0]; num_pos[b] = redi[0]; }
}

// ---------------------------------------------------------------------------
// Kernel 2: total positive count (needed for the gather_tgt[0] quirk and N)
// ---------------------------------------------------------------------------
__global__ __launch_bounds__(128) void totpos_kernel(
    const int* __restrict__ num_pos, int* __restrict__ totpos, int B)
{
    __shared__ int red[128];
    int tid = threadIdx.x, s = 0;
    for (int j = tid; j < B; j += 128) s += num_pos[j];
    red[tid] = s; __syncthreads();
    for (int st = 64; st > 0; st >>= 1) {
        if (tid < st) red[tid] += red[tid + st];
        __syncthreads();
    }
    if (tid == 0) totpos[0] = red[0];
}

// ---------------------------------------------------------------------------
// Kernel 3: logsumexp over C=21 classes + ranking loss (one pass over 94 MB)
// ---------------------------------------------------------------------------
__global__ __launch_bounds__(256) void rank_kernel(
    const float* __restrict__ conf, const int* __restrict__ conf_t,
    const int* __restrict__ totpos, float* __restrict__ rank,
    float* __restrict__ lsea, int BP, int C)
{
    int i = blockIdx.x * 256 + threadIdx.x;
    if (i >= BP) return;
    const float* x = conf + (size_t)i * C;
    float m = x[0];
    for (int c = 1; c < C; ++c) m = fmaxf(m, x[c]);
    float s = 0.0f;
    for (int c = 0; c < C; ++c) s += expf(x[c] - m);
    float l = m + logf(s);
    lsea[i] = l;
    int ct = conf_t[i];
    int g = (ct != 0) ? 1 : 0;
    if (i == 0 && totpos[0] > 0) g = 1;          // torch nonzero() quirk
    float rl = (ct > 0) ? 0.0f : (l - x[g]);     // rl >= +0 provably (s>=1, m>=x[g] when max)
    rank[i] = rl;
}

// ---------------------------------------------------------------------------
// Kernel 4: exact k-th-largest via 4-pass MSB radix select (per batch).
// rank values are nonnegative floats -> uint bit order == value order.
// ---------------------------------------------------------------------------
__global__ __launch_bounds__(256) void select_kernel(
    const float* __restrict__ rank, const int* __restrict__ num_pos,
    unsigned* __restrict__ thr, int* __restrict__ slots, int P)
{
    const int b = blockIdx.x, tid = threadIdx.x;
    __shared__ unsigned s_hist[256];
    __shared__ unsigned s_prefix;
    __shared__ unsigned s_rem;
    const size_t bP = (size_t)b * P;

    int k = 3 * num_pos[b];
    if (k > P - 1) k = P - 1;
    if (k <= 0) {
        if (tid == 0) { thr[b] = 0x7f800000u; slots[b] = 0; } // +inf => empty neg set
        return;
    }
    if (tid == 0) { s_prefix = 0u; s_rem = (unsigned)k; }
    __syncthreads();

    for (int pass = 0; pass < 4; ++pass) {
        int shift = 24 - 8 * pass;
        unsigned pmask = (pass == 0) ? 0u : (0xFFFFFFFFu << (shift + 8));
        s_hist[tid] = 0u;
        __syncthreads();
        unsigned pref = s_prefix;
        for (int p = tid; p < P; p += 256) {
            unsigned v = __float_as_uint(rank[bP + p]);
            if ((v & pmask) == pref)
                atomicAdd(&s_hist[(v >> shift) & 0xffu], 1u);
        }
        __syncthreads();
        if (tid == 0) {
            unsigned cum = 0; int d = 255;
            for (; d > 0; --d) {
                if (cum + s_hist[d] >= s_rem) break;
                cum += s_hist[d];
            }
            s_rem -= cum;
            s_prefix = pref | ((unsigned)d << shift);
        }
        __syncthreads();
    }
    if (tid == 0) { thr[b] = s_prefix; slots[b] = (int)s_rem; }
}

// ---------------------------------------------------------------------------
// Kernel 5: cross-entropy over pos|neg. Stable-argsort tie order reproduced
// with an ordered prefix scan over elements equal to the threshold value.
// ---------------------------------------------------------------------------
__global__ __launch_bounds__(256) void ce_kernel(
    const float* __restrict__ conf, const int* __restrict__ conf_t,
    const float* __restrict__ rank, const float* __restrict__ lsea,
    const unsigned* __restrict__ thr, const int* __restrict__ slots,
    float* __restrict__ lossc_part, int P, int C)
{
    const int b = blockIdx.x, tid = threadIdx.x;
    __shared__ int   scan[256];
    __shared__ float redf[256];
    const size_t bP = (size_t)b * P;
    const unsigned thrBits = thr[b];
    const int slotsEq = slots[b];

    float acc = 0.0f;
    int base = 0;
    for (int start = 0; start < P; start += 256) {
        int p = start + tid;
        int valid = (p < P);
        unsigned vb = 0; int isEq = 0;
        if (valid) {
            vb = __float_as_uint(rank[bP + p]);
            isEq = (vb == thrBits) ? 1 : 0;
        }
        scan[tid] = isEq;
        __syncthreads();
        for (int off = 1; off < 256; off <<= 1) {
            int t = (tid >= off) ? scan[tid - off] : 0;
            __syncthreads();
            scan[tid] += t;
            __syncthreads();
        }
        int excl = scan[tid] - isEq;
        int chunkTot = scan[255];
        if (valid) {
            bool neg = (vb > thrBits) || (isEq && (base + excl) < slotsEq);
            int ct = conf_t[bP + p];
            bool pos = (ct > 0);
            if (pos || neg)
                acc += lsea[bP + p] - conf[(bP + p) * (size_t)C + ct];
        }
        base += chunkTot;
        __syncthreads();
    }
    redf[tid] = acc;
    __syncthreads();
    for (int s = 128; s > 0; s >>= 1) {
        if (tid < s) redf[tid] += redf[tid + s];
        __syncthreads();
    }
    if (tid == 0) lossc_part[b] = redf[0];
}

// ---------------------------------------------------------------------------
// Kernel 6: finalize. Single wave32. The 128-way partial reduction uses
// V_WMMA_F32_16X16X4_F32 with an all-ones A: D[0,n] = sum_k B[k,n], so
// sum_{n} D[0,n] (row M=0 = VGPR0, lanes 0..15) is the grand total of all
// 64 B slots regardless of B's lane layout (bijection). One instruction
// performs the 4-way k sums; 4 xor-shuffles add the 16 columns.
// ---------------------------------------------------------------------------
__device__ inline float wave_sum_wmma(const float* __restrict__ arr, int B)
{
    int lane = threadIdx.x;
    float s0 = 0.0f, s1 = 0.0f;
    for (int j = lane;      j < B; j += 64) s0 += arr[j];
    for (int j = lane + 32; j < B; j += 64) s1 += arr[j];
    v2f a;  a[0] = 1.0f; a[1] = 1.0f;   // A all-ones (any layout)
    v2f bb; bb[0] = s0;  bb[1] = s1;    // 64 slots, bijective lane mapping
    v8f c = {};
    v8f d = __builtin_amdgcn_wmma_f32_16x16x4_f32(
        false, a, false, bb, (short)0, c, false, false);
    float v = d[0];                     // D row M=0 across lanes 0..15
    v += __shfl_xor(v, 8, 16);
    v += __shfl_xor(v, 4, 16);
    v += __shfl_xor(v, 2, 16);
    v += __shfl_xor(v, 1, 16);
    return v;                           // lane 0 holds the grand total
}

__global__ __launch_bounds__(32) void finalize_kernel(
    const float* __restrict__ lossl_part, const float* __restrict__ lossc_part,
    const int* __restrict__ totpos, float* __restrict__ out, int B)
{
    float ll = wave_sum_wmma(lossl_part, B);
    float lc = wave_sum_wmma(lossc_part, B);
    if (threadIdx.x == 0) {
        float N = (float)totpos[0];
        out[0] = ll / N;
        out[1] = lc / N;
    }
}

// ---------------------------------------------------------------------------
extern "C" void kernel_launch(void* const* d_in, const int* in_sizes, int n_in,
                              void* d_out, int out_size, void* d_ws, size_t ws_size,
                              hipStream_t stream)
{
    (void)n_in; (void)out_size; (void)ws_size;
    const float* loc     = (const float*)d_in[0];
    const float* conf    = (const float*)d_in[1];
    const float* priors  = (const float*)d_in[2];
    const float* targets = (const float*)d_in[3];

    const int P = in_sizes[2] / 4;
    const int B = in_sizes[0] / (P * 4);
    const int C = in_sizes[1] / (B * P);
    int O = in_sizes[3] / (B * 5);
    if (O > MAXO) O = MAXO;
    const size_t BP = (size_t)B * P;

    char* w = (char*)d_ws;
    int*      conf_t = (int*)w;        w += BP * sizeof(int);
    float*    rank   = (float*)w;      w += BP * sizeof(float);
    float*    lsea   = (float*)w;      w += BP * sizeof(float);
    int*      npos   = (int*)w;        w += B * sizeof(int);
    float*    lossl  = (float*)w;      w += B * sizeof(float);
    float*    lossc  = (float*)w;      w += B * sizeof(float);
    unsigned* thr    = (unsigned*)w;   w += B * sizeof(unsigned);
    int*      slots  = (int*)w;        w += B * sizeof(int);
    int*      totpos = (int*)w;

    match_kernel<<<B, 256, 0, stream>>>(loc, priors, targets, conf_t, npos, lossl, P, O);
    totpos_kernel<<<1, 128, 0, stream>>>(npos, totpos, B);
    const int nb = (int)((BP + 255) / 256);
    rank_kernel<<<nb, 256, 0, stream>>>(conf, conf_t, totpos, rank, lsea, (int)BP, C);
    select_kernel<<<B, 256, 0, stream>>>(rank, npos, thr, slots, P);
    ce_kernel<<<B, 256, 0, stream>>>(conf, conf_t, rank, lsea, thr, slots, lossc, P, C);
    finalize_kernel<<<1, 32, 0, stream>>>(lossl, lossc, totpos, (float*)d_out, B);
}